// OnlineAverager_60215441490398
// MI455X (gfx1250) — compile-verified
//
#include <hip/hip_runtime.h>

// ---------------- problem constants (from reference) ----------------
#define UPDATE_SIZE   4096
#define BATCH_SIZE    256
#define NUM_UPDATES   16
#define SNAPSHOT_SIZE (UPDATE_SIZE * NUM_UPDATES)                    // 65536
#define OUT_SIZE      ((long long)UPDATE_SIZE * BATCH_SIZE)          // 1048576
#define SNAP_LEN      (SNAPSHOT_SIZE + (long long)(BATCH_SIZE - 1) * UPDATE_SIZE) // 1110016
#define NUM_B_BLOCKS  271                                            // SNAP_LEN / UPDATE_SIZE
#define COLSPLIT      4                                              // col-groups per b-block
#define COLS_PER_BLK  (UPDATE_SIZE / COLSPLIT)                       // 1024
#define LDS_ROW_DW    (COLS_PER_BLK + 16)  // 1040 dw row stride: +64B pad -> bank-conflict-free

typedef float v2f __attribute__((ext_vector_type(2)));
typedef float v8f __attribute__((ext_vector_type(8)));
typedef int   v4i __attribute__((vector_size(4 * sizeof(int))));

#define AS1 __attribute__((address_space(1)))
#define AS3 __attribute__((address_space(3)))

// Async global->LDS copy of 16B per lane (GLOBAL_LOAD_ASYNC_TO_LDS_B128).
__device__ __forceinline__ void async_copy_b128(const float* gsrc, float* ldst_generic) {
#if __has_builtin(__builtin_amdgcn_global_load_async_to_lds_b128)
    __builtin_amdgcn_global_load_async_to_lds_b128(
        (AS1 v4i*)gsrc, (AS3 v4i*)ldst_generic, 0, 0);
#else
    unsigned ldsAddr = (unsigned)(size_t)(AS3 float*)ldst_generic;
    asm volatile("global_load_async_to_lds_b128 %0, %1, off"
                 :: "v"(ldsAddr), "v"(gsrc) : "memory");
#endif
}

__device__ __forceinline__ void wait_async_zero() {
#if __has_builtin(__builtin_amdgcn_s_wait_asynccnt)
    __builtin_amdgcn_s_wait_asynccnt(0);
    asm volatile("" ::: "memory");
#else
    asm volatile("s_wait_asynccnt 0x0" ::: "memory");
#endif
}

// One workgroup = 256 threads = 8 wave32; handles 1024 columns of one b-block.
// Stage 16 diagonal update rows into LDS via async B128 DMA, then each wave:
// 8 chains x 16 cols, each chain = 4x V_WMMA_F32_16X16X4_F32 fed from LDS.
// Thread t stages float4 #t of each row -> wave wv stages exactly the columns
// it consumes, so a per-wave s_wait_asynccnt(0) suffices (no block barrier).
__global__ void __launch_bounds__(256)
oa_main_kernel(const float* __restrict__ update,
               const float* __restrict__ snapshot,
               const float* __restrict__ uidxp,
               float* __restrict__ out)
{
    __shared__ float ldsX[NUM_UPDATES * LDS_ROW_DW];   // 66,560 B

    const int bid  = blockIdx.x;
    const int b    = bid >> 2;                 // output block index 0..270
    const int colBase = (bid & (COLSPLIT - 1)) * COLS_PER_BLK;
    const int t    = threadIdx.x;
    const int lane = t & 31;
    const int wv   = t >> 5;
    const int n    = lane & 15;                // B/D layout: N = lane%16
    const bool hiHalf = lane >= 16;            // lanes 16..31: K=2,3 (A/B), M=8.. (D)

    const float uidx = uidxp[0];

    // ---- per-b scan coefficients (uniform suffix-product over touches) ----
    float cs[16];
    float suffix = 1.0f;
#pragma unroll
    for (int s = 0; s < 16; ++s) {
        const int i = b - s;                   // ascending s == descending update i
        float c = 0.0f;
        if (i >= 0 && i < BATCH_SIZE) {
            const float nrm = (float)(NUM_UPDATES - s);
            const float hi  = uidx + (float)i + 1.0f;
            const float w   = fminf(hi, fmaxf(nrm, 0.0f));  // jnp.clip(norm, 0, hi)
            const float r   = 1.0f / w;
            c = r * suffix;
            suffix *= (1.0f - r);
        }
        cs[s] = c;
    }
    const float msnap = suffix;

    // ---- stage the 16 diagonal rows (clamped rows carry coef 0) ----
#pragma unroll
    for (int s = 0; s < 16; ++s) {
        int i = b - s;
        i = i < 0 ? 0 : (i > BATCH_SIZE - 1 ? BATCH_SIZE - 1 : i);
        const float* g = update + (size_t)i * SNAPSHOT_SIZE + (size_t)s * UPDATE_SIZE
                                + colBase + 4 * t;
        async_copy_b128(g, &ldsX[s * LDS_ROW_DW + 4 * t]);
    }
    wait_async_zero();

    // ---- compute: 8 chains of 16 columns per wave ----
    const float* snapRow = snapshot + (size_t)b * UPDATE_SIZE + colBase;
    float*       outRow  = out      + (size_t)b * UPDATE_SIZE + colBase;
    const int myCol = wv * 128;
    const int kOff  = hiHalf ? 2 : 0;          // half-wave K offset

    for (int ch = 0; ch < 8; ++ch) {
        const int c0 = myCol + ch * 16;

        // C accumulator: all 16 rows identical = msnap * snapshot[p]
        const float sv = msnap * __builtin_nontemporal_load(snapRow + c0 + n);
        v8f acc = {sv, sv, sv, sv, sv, sv, sv, sv};

#pragma unroll
        for (int j = 0; j < 4; ++j) {
            // A (16x4 f32): rows replicated -> A[m][k] = cs[4j+k]
            v2f a;
            a.x = hiHalf ? cs[4 * j + 2] : cs[4 * j + 0];
            a.y = hiHalf ? cs[4 * j + 3] : cs[4 * j + 1];

            // B (4x16 f32) from LDS: B[k][n] = X[4j+k][c0+n]
            const int sA = 4 * j + kOff;
            v2f bb;
            bb.x = ldsX[sA * LDS_ROW_DW + c0 + n];
            bb.y = ldsX[(sA + 1) * LDS_ROW_DW + c0 + n];

            acc = __builtin_amdgcn_wmma_f32_16x16x4_f32(
                false, a, false, bb, (short)0, acc, false, false);
        }

        // D rows identical; lane L<16 VGPR0 holds (M=0, N=L) == out[c0+L]
        if (!hiHalf) {
            __builtin_nontemporal_store(acc[0], outRow + c0 + lane);
        }
    }
}

// Zero-fill new_snapshot tail [SNAP_LEN, SNAP_LEN+OUT_SIZE) and write scalar.
__global__ void __launch_bounds__(256)
oa_tail_kernel(float* __restrict__ out, const float* __restrict__ uidxp)
{
    const size_t tid = (size_t)blockIdx.x * blockDim.x + threadIdx.x;
    float4* p = reinterpret_cast<float4*>(out + SNAP_LEN);
    p[tid] = make_float4(0.0f, 0.0f, 0.0f, 0.0f);
    if (tid == 0) {
        out[OUT_SIZE + SNAP_LEN] = uidxp[0] + (float)BATCH_SIZE;
    }
}

extern "C" void kernel_launch(void* const* d_in, const int* in_sizes, int n_in,
                              void* d_out, int out_size, void* d_ws, size_t ws_size,
                              hipStream_t stream)
{
    (void)in_sizes; (void)n_in; (void)out_size; (void)d_ws; (void)ws_size;

    const float* update    = (const float*)d_in[0];
    const float* snapshot  = (const float*)d_in[1];
    const float* uidx      = (const float*)d_in[2];
    float* out             = (float*)d_out;

    dim3 blk(256);
    // main: 271 b-blocks x 4 column segments
    oa_main_kernel<<<dim3(NUM_B_BLOCKS * COLSPLIT), blk, 0, stream>>>(
        update, snapshot, uidx, out);
    // tail: OUT_SIZE zeros as float4 -> 1048576/4/256 = 1024 blocks
    oa_tail_kernel<<<dim3(1024), blk, 0, stream>>>(out, uidx);
}